// SpatialGather_Module_38611755991120
// MI455X (gfx1250) — compile-verified
//
#include <hip/hip_runtime.h>

#define B_      16
#define K_      29
#define KM_     32
#define C_      512
#define S_      16384      // 128*128
#define SBLK    32         // s-values per LDS stage (double-buffered)
#define ROWP    36         // LDS row stride (floats): conflict-free ds_load_b64
#define CCHUNK  128        // C columns per block (8 waves x 16)
#define SCHUNKS 8
#define SCHUNK  (S_ / SCHUNKS)   // 2048
#define NBLK    (SCHUNK / SBLK)  // 64 s-blocks per workgroup

typedef float v2f __attribute__((ext_vector_type(2)));
typedef float v8f __attribute__((ext_vector_type(8)));

__global__ void zero_out_kernel(float* __restrict__ out, int n) {
    int i = blockIdx.x * blockDim.x + threadIdx.x;
    if (i < n) out[i] = 0.0f;
}

// One block per (b,k) row: compute row max and 1/sum(exp(x-max)) over S_.
__global__ __launch_bounds__(256) void softmax_stats_kernel(const float* __restrict__ probs,
                                                            float* __restrict__ ws) {
    __shared__ float red[256];
    const int row = blockIdx.x;            // b*K_ + k
    const int t   = threadIdx.x;
    const float4* p4 = (const float4*)(probs + (size_t)row * S_);

    float m = -3.402823466e+38f;
    for (int i = t; i < S_ / 4; i += 256) {
        float4 v = p4[i];
        m = fmaxf(m, fmaxf(fmaxf(v.x, v.y), fmaxf(v.z, v.w)));
    }
    red[t] = m; __syncthreads();
    for (int s = 128; s > 0; s >>= 1) {
        if (t < s) red[t] = fmaxf(red[t], red[t + s]);
        __syncthreads();
    }
    const float mx = red[0]; __syncthreads();

    float acc = 0.0f;
    for (int i = t; i < S_ / 4; i += 256) {
        float4 v = p4[i];
        acc += __expf(v.x - mx) + __expf(v.y - mx) + __expf(v.z - mx) + __expf(v.w - mx);
    }
    red[t] = acc; __syncthreads();
    for (int s = 128; s > 0; s >>= 1) {
        if (t < s) red[t] += red[t + s];
        __syncthreads();
    }
    if (t == 0) { ws[row] = mx; ws[B_ * K_ + row] = 1.0f / red[0]; }
}

// CDNA5 async memory->LDS copy (16B per lane), tracked by ASYNCcnt.
__device__ __forceinline__ void async_ld_b128(unsigned lds_byte_addr, const float* gaddr) {
    asm volatile("global_load_async_to_lds_b128 %0, %1, off"
                 :: "v"(lds_byte_addr), "v"(gaddr) : "memory");
}
__device__ __forceinline__ void wait_async0() {
    asm volatile("s_wait_asynccnt 0x0" ::: "memory");
}

// GEMM: out[b,c,k] += sum_s attn[b,k,s] * feats[b,c,s] via v_wmma_f32_16x16x4_f32,
// feats staged through LDS with global_load_async_to_lds_b128, double-buffered.
__global__ __launch_bounds__(256) void gather_wmma_kernel(const float* __restrict__ feats,
                                                          const float* __restrict__ probs,
                                                          const float* __restrict__ ws,
                                                          float* __restrict__ out) {
    __shared__ float lds_a[2 * KM_ * ROWP];      // 2 x 32 x (32+pad) attn tiles
    __shared__ float lds_b[2 * CCHUNK * ROWP];   // 2 x 128 x (32+pad) feats tiles

    const int bid = blockIdx.x;
    const int b   = bid >> 5;                // 16 batches
    const int cc  = bid & 3;                 // 4 C-chunks
    const int sc  = (bid >> 2) & 7;          // 8 S-chunks
    const int c0  = cc * CCHUNK;
    const int s_begin = sc * SCHUNK;

    const int t    = threadIdx.x;
    const int lane = t & 31;
    const int wv   = t >> 5;                 // wave 0..7 -> N-tile
    const int n    = lane & 15;              // M/N index within tile
    const int kh   = lane >> 4;              // K-half of the fragment

    // A-staging constants: thread t stages row am (0..31), float4 slot asl
    const int am  = t >> 3;                  // 32 rows x 8 float4 = 256 threads
    const int asl = (t & 7) * 4;             // s-local float offset
    const float mxA = (am < K_) ? ws[b * K_ + am] : 0.0f;
    const float isA = (am < K_) ? ws[B_ * K_ + b * K_ + am] : 0.0f;
    const size_t aprow = (size_t)b * K_ * S_ + (size_t)am * S_;

    // B async-staging constants: thread t copies float4 slots t + i*256
    const size_t feats_base = (size_t)b * C_ * S_;
    const unsigned bbase = (unsigned)(size_t)(&lds_b[0]);   // LDS byte offset of lds_b

    v8f acc0 = {};   // M rows 0..15
    v8f acc1 = {};   // M rows 16..31

    // ---------------- staging helpers (inlined manually) ----------------
    // stage A tile for buffer `buf` at global s-offset s0
    auto stage_a = [&](int buf, int s0) {
        float4 v = make_float4(0.f, 0.f, 0.f, 0.f);
        if (am < K_) {
            const float4 p = *(const float4*)(probs + aprow + s0 + asl);
            v.x = __expf(p.x - mxA) * isA;
            v.y = __expf(p.y - mxA) * isA;
            v.z = __expf(p.z - mxA) * isA;
            v.w = __expf(p.w - mxA) * isA;
        }
        *(float4*)&lds_a[buf * (KM_ * ROWP) + am * ROWP + asl] = v;
    };
    // issue async copies of the feats tile for buffer `buf` at s-offset s0
    auto stage_b = [&](int buf, int s0) {
        #pragma unroll
        for (int i = 0; i < 4; ++i) {
            int idx = t + i * 256;           // float4 index 0..1023
            int cl  = idx >> 3;              // 0..127
            int sl  = (idx & 7) * 4;         // float offset within row
            unsigned lad = bbase + (unsigned)(buf * (CCHUNK * ROWP) + cl * ROWP + sl) * 4u;
            async_ld_b128(lad, feats + feats_base + (size_t)(c0 + cl) * S_ + s0 + sl);
        }
    };

    // ---------------- pipeline ----------------
    stage_a(0, s_begin);
    stage_b(0, s_begin);
    wait_async0();
    __syncthreads();

    for (int i = 0; i < NBLK; ++i) {
        const int cur = i & 1;
        const int s0  = s_begin + i * SBLK;
        if (i + 1 < NBLK) {                  // stage next block into other buffer
            stage_a(cur ^ 1, s0 + SBLK);
            stage_b(cur ^ 1, s0 + SBLK);
        }
        // compute on current buffer: 8 k-steps of 4 -> 16 WMMA
        const float* ap0 = &lds_a[cur * (KM_ * ROWP) + n * ROWP + 2 * kh];
        const float* ap1 = ap0 + 16 * ROWP;
        const float* bp  = &lds_b[cur * (CCHUNK * ROWP) + (wv * 16 + n) * ROWP + 2 * kh];
        #pragma unroll
        for (int kk = 0; kk < SBLK / 4; ++kk) {
            v2f a0 = *(const v2f*)(ap0 + kk * 4);
            v2f a1 = *(const v2f*)(ap1 + kk * 4);
            v2f bb = *(const v2f*)(bp + kk * 4);
            acc0 = __builtin_amdgcn_wmma_f32_16x16x4_f32(false, a0, false, bb,
                                                         (short)0, acc0, false, false);
            acc1 = __builtin_amdgcn_wmma_f32_16x16x4_f32(false, a1, false, bb,
                                                         (short)0, acc1, false, false);
        }
        wait_async0();                       // next block's async copies have landed
        __syncthreads();                     // all waves done staging + computing
    }

    // ---------------- scatter partials: out (B, C, K, 1) -> b*C*K + c*K + m
    const int c = c0 + wv * 16 + n;
    float* obase = out + ((size_t)b * C_ + c) * K_;
    #pragma unroll
    for (int i = 0; i < 8; ++i) {
        int m = i + kh * 8;                  // 0..15, always < K_
        atomicAdd(obase + m, acc0[i]);
        int m2 = m + 16;                     // 16..31
        if (m2 < K_) atomicAdd(obase + m2, acc1[i]);
    }
}

extern "C" void kernel_launch(void* const* d_in, const int* in_sizes, int n_in,
                              void* d_out, int out_size, void* d_ws, size_t ws_size,
                              hipStream_t stream) {
    const float* feats = (const float*)d_in[0];   // (16, 512, 128, 128) f32
    const float* probs = (const float*)d_in[1];   // (16, 29, 128, 128) f32
    float* out = (float*)d_out;                   // (16, 512, 29, 1) f32
    float* ws  = (float*)d_ws;                    // 464 maxes + 464 inv-sums

    const int n_out = B_ * C_ * K_;
    zero_out_kernel<<<(n_out + 255) / 256, 256, 0, stream>>>(out, n_out);
    softmax_stats_kernel<<<B_ * K_, 256, 0, stream>>>(probs, ws);
    gather_wmma_kernel<<<B_ * 4 * SCHUNKS, 256, 0, stream>>>(feats, probs, ws, out);
}